// WeightedRadialAEVComputer_84335977825045
// MI455X (gfx1250) — compile-verified
//
#include <hip/hip_runtime.h>

// ---------------------------------------------------------------------------
// WeightedRadialAEV on MI455X (gfx1250)
//   GR[b,i,p] = sum_j z[b,j] * fc(d[b,i,j]) * exp(-16*(d[b,i,j]-shf_p)^2)
//   mask: (d < 5.2) && (d != 0)
// B=16, N=512, P=16.  Transcendental-bound (67M v_exp); data fed via TDM
// tensor_load_to_lds into LDS, compute from LDS, wave32 butterfly reduce.
// ---------------------------------------------------------------------------

#define NATOMS 512
#define NFEAT  16
#define ROWS_PER_BLOCK 8   // 8 waves of 32 -> 256 threads, one wave per row

typedef __attribute__((ext_vector_type(4))) unsigned int v4u;
typedef __attribute__((ext_vector_type(4))) int          v4i;
typedef __attribute__((ext_vector_type(8))) int          v8i;

// Issue one TDM 2D tile load: tile_rows x row_elems f32 elements, row stride
// row_stride_elems (elements), from gaddr into LDS byte offset lds_byte.
__device__ __forceinline__ void tdm_load_2d(unsigned lds_byte, const void* gaddr,
                                            unsigned tile_rows, unsigned row_elems,
                                            unsigned row_stride_elems) {
  unsigned long long ga = (unsigned long long)(uintptr_t)gaddr;

  // D# group 0: count=1 | lds_addr | global_addr[56:0] | type=2 ("image")
  v4u g0;
  g0.x = 1u;
  g0.y = lds_byte;
  g0.z = (unsigned)ga;
  g0.w = (unsigned)((ga >> 32) & 0x01FFFFFFu) | (2u << 30);

  // D# group 1: wg_mask=0, data_size=4B, tensor_dim0/1, tile_dim0/1/2, stride0
  v8i g1;
  g1[0] = (int)(2u << 16);                                        // data_size=2 -> 4 bytes
  g1[1] = (int)((row_elems & 0xFFFFu) << 16);                     // tensor_dim0[15:0]
  g1[2] = (int)(((row_elems >> 16) & 0xFFFFu) |
                ((tile_rows & 0xFFFFu) << 16));                   // dim0 hi | tensor_dim1 lo
  g1[3] = (int)(((tile_rows >> 16) & 0xFFFFu) |
                ((row_elems & 0xFFFFu) << 16));                   // dim1 hi | tile_dim0
  g1[4] = (int)(tile_rows & 0xFFFFu);                             // tile_dim1 (tile_dim2=0)
  g1[5] = (int)row_stride_elems;                                  // tensor_dim0_stride lo
  g1[6] = 0;
  g1[7] = 0;

  v4i z4 = {0, 0, 0, 0};
#if __has_include(<hip/amd_detail/amd_gfx1250_TDM.h>)
  v8i z8 = {0, 0, 0, 0, 0, 0, 0, 0};
  __builtin_amdgcn_tensor_load_to_lds(g0, g1, z4, z4, z8, 0);     // 6-arg toolchain
#else
  __builtin_amdgcn_tensor_load_to_lds(g0, g1, z4, z4, 0);         // 5-arg ROCm 7.2
#endif
}

__global__ __launch_bounds__(ROWS_PER_BLOCK * 32)
void aev_radial_kernel(const float* __restrict__ d,   // [16][512][512]
                       const float* __restrict__ z,   // [16][512]
                       float* __restrict__ out) {     // [16][512][16]
  __shared__ float smem_d[ROWS_PER_BLOCK * NATOMS];   // 16 KB tile
  __shared__ float smem_z[NATOMS];                    // 2 KB row

  const int lane = threadIdx.x & 31;
  const int wave = threadIdx.x >> 5;
  const int blk  = blockIdx.x;        // 0..1023
  const int b    = blk >> 6;          // batch
  const int i0   = (blk & 63) * ROWS_PER_BLOCK;

  if (threadIdx.x < 32) {             // wave 0: blockIdx-uniform descriptors
    const float* dtile = d + ((size_t)(b * NATOMS + i0)) * NATOMS;
    tdm_load_2d((unsigned)(uintptr_t)&smem_d[0], dtile,
                ROWS_PER_BLOCK, NATOMS, NATOMS);
    tdm_load_2d((unsigned)(uintptr_t)&smem_z[0], z + (size_t)b * NATOMS,
                1, NATOMS, NATOMS);
    __builtin_amdgcn_s_wait_tensorcnt(0);
  }
  __syncthreads();

  float acc[NFEAT];
#pragma unroll
  for (int p = 0; p < NFEAT; ++p) acc[p] = 0.0f;

  const float* drow = &smem_d[wave * NATOMS];
  const float kcos = 0.09615384615384616f;   // 1/(2*RCR): v_cos takes revolutions
  const float kexp = -23.083120654223414f;   // -ETA * log2(e)

  for (int j = lane; j < NATOMS; j += 32) {
    float dv = drow[j];
    bool  m  = (dv < 5.2f) && (dv != 0.0f);
    float fc = 0.5f * __builtin_amdgcn_cosf(dv * kcos) + 0.5f;
    float wz = m ? smem_z[j] * fc : 0.0f;
#pragma unroll
    for (int p = 0; p < NFEAT; ++p) {
      float t = dv - (0.9f + 0.26875f * (float)p);
      float g = __builtin_amdgcn_exp2f(kexp * t * t);  // exp(-16*t^2)
      acc[p] = fmaf(wz, g, acc[p]);
    }
  }

  // wave32 butterfly: every lane ends with the full j-sum per feature
#pragma unroll
  for (int p = 0; p < NFEAT; ++p) {
    float v = acc[p];
    v += __shfl_xor(v, 16, 32);
    v += __shfl_xor(v, 8, 32);
    v += __shfl_xor(v, 4, 32);
    v += __shfl_xor(v, 2, 32);
    v += __shfl_xor(v, 1, 32);
    acc[p] = v;
  }

  float ov = acc[0];
#pragma unroll
  for (int p = 1; p < NFEAT; ++p) ov = (lane == p) ? acc[p] : ov;
  if (lane < NFEAT) {
    const int row = b * NATOMS + i0 + wave;
    out[(size_t)row * NFEAT + lane] = ov;
  }
}

extern "C" void kernel_launch(void* const* d_in, const int* in_sizes, int n_in,
                              void* d_out, int out_size, void* d_ws, size_t ws_size,
                              hipStream_t stream) {
  const float* d  = (const float*)d_in[0];   // distance_matrices [16,512,512]
  const float* z  = (const float*)d_in[1];   // atomic_numbers_batch [16,512]
  float* out      = (float*)d_out;           // [16,512,16]
  (void)in_sizes; (void)n_in; (void)out_size; (void)d_ws; (void)ws_size;

  dim3 grid(16 * (NATOMS / ROWS_PER_BLOCK));  // 1024 blocks
  dim3 block(ROWS_PER_BLOCK * 32);            // 256 threads = 8 waves
  aev_radial_kernel<<<grid, block, 0, stream>>>(d, z, out);
}